// HybridGATGCN_37091337568515
// MI455X (gfx1250) — compile-verified
//
#include <hip/hip_runtime.h>
#include <math.h>

#define N_NODES 100000
#define N_EDGES 1600000
#define ETOT    (N_EDGES + N_NODES)   // edges + self loops
#define IN_F    128
#define HID_F   32
#define OUT_F   16
#define HEADS   2
#define EPS_BN  1e-5f
#define NEG_SLOPE 0.2f

typedef __attribute__((ext_vector_type(2))) float v2f;
typedef __attribute__((ext_vector_type(8))) float v8f;

// ---------------------------------------------------------------------------
// fp32 GEMM on the WMMA pipe: C[M,NCOLS] = A[M,K] @ B[K,NCOLS]
// B (<=16KB) is staged into LDS once per block; each wave owns one 16-row
// stripe and computes ALL NCOLS/16 column tiles, reusing its A fragment
// across COLT WMMAs per k-step (cuts A HBM traffic by COLT x).
// Fragment layouts per ISA 7.12.2:
//   A 16x4 : lanes 0-15 row M=l hold K=k0,k0+1 ; lanes 16-31 hold K=k0+2,k0+3
//   B 4x16 : mirrored ; C : VGPR r -> row r (lanes 0-15) / row 8+r (lanes 16-31)
// M = N_NODES = 6250*16 exactly, so no row tail; EXEC stays all-1s in WMMA.
// ---------------------------------------------------------------------------
template <int K, int NCOLS>
__global__ __launch_bounds__(256) void wmma_gemm(const float* __restrict__ A,
                                                 const float* __restrict__ B,
                                                 float* __restrict__ C) {
    constexpr int COLT = NCOLS / 16;
    __shared__ float Bs[K * NCOLS];
    // cooperative stage of the whole weight matrix into LDS
    for (int i = threadIdx.x; i < K * NCOLS; i += 256) Bs[i] = B[i];
    __syncthreads();

    const int wave = (blockIdx.x * blockDim.x + threadIdx.x) >> 5;  // row tile
    const int m0 = wave * 16;
    if (m0 >= N_NODES) return;          // uniform per wave; after barrier

    const int lane = threadIdx.x & 31;
    const int half = lane >> 4;         // 0: K k0,k0+1   1: K k0+2,k0+3
    const int l    = lane & 15;

    v8f c[COLT];
    #pragma unroll
    for (int t = 0; t < COLT; ++t) c[t] = (v8f){};

    #pragma unroll
    for (int k0 = 0; k0 < K; k0 += 4) {
        v2f a = *(const v2f*)(A + (size_t)(m0 + l) * K + k0 + 2 * half);
        #pragma unroll
        for (int t = 0; t < COLT; ++t) {
            const float* bp = Bs + (k0 + 2 * half) * NCOLS + t * 16 + l;
            v2f b; b.x = bp[0]; b.y = bp[NCOLS];
            c[t] = __builtin_amdgcn_wmma_f32_16x16x4_f32(false, a, false, b,
                                                         (short)0, c[t], false, false);
        }
    }
    #pragma unroll
    for (int t = 0; t < COLT; ++t) {
        float* cp = C + (size_t)(m0 + half * 8) * NCOLS + t * 16 + l;
        #pragma unroll
        for (int r = 0; r < 8; ++r) cp[(size_t)r * NCOLS] = c[t][r];
    }
}

// ---------------------------------------------------------------------------
// Elementwise / scatter helpers
// ---------------------------------------------------------------------------
__global__ void fill_kernel(float* __restrict__ p, float v, long n) {
    long i = (long)blockIdx.x * blockDim.x + threadIdx.x;
    if (i < n) p[i] = v;
}

__global__ void deg_kernel(const int* __restrict__ dst, float* __restrict__ deg) {
    int e = blockIdx.x * blockDim.x + threadIdx.x;
    if (e < N_EDGES) atomicAdd(&deg[dst[e]], 1.0f);
}

__global__ void rsqrt_kernel(float* __restrict__ d) {
    int n = blockIdx.x * blockDim.x + threadIdx.x;
    if (n < N_NODES) d[n] = rsqrtf(fmaxf(d[n], 1.0f));
}

// out[dst] += H[src] * dinv[src]*dinv[dst]   (edge range includes self loops)
__global__ void gcn_scatter(const float* __restrict__ H, const int* __restrict__ src,
                            const int* __restrict__ dst, const float* __restrict__ dinv,
                            float* __restrict__ out, int F) {
    long idx = (long)blockIdx.x * blockDim.x + threadIdx.x;
    int groups = F >> 2;
    long e = idx / groups;
    if (e >= ETOT) return;
    int fg = (int)(idx % groups) << 2;
    int s, d;
    if (e < N_EDGES) { s = src[e]; d = dst[e]; } else { s = d = (int)(e - N_EDGES); }
    float norm = dinv[s] * dinv[d];
    const float* hs = H + (size_t)s * F + fg;
    float* od = out + (size_t)d * F + fg;
    atomicAdd(&od[0], hs[0] * norm);
    atomicAdd(&od[1], hs[1] * norm);
    atomicAdd(&od[2], hs[2] * norm);
    atomicAdd(&od[3], hs[3] * norm);
}

__global__ void bias_bn_relu(float* __restrict__ h, const float* __restrict__ b,
                             const float* __restrict__ g, const float* __restrict__ beta,
                             const float* __restrict__ m, const float* __restrict__ v) {
    long i = (long)blockIdx.x * blockDim.x + threadIdx.x;
    if (i >= (long)N_NODES * HID_F) return;
    int f = (int)(i % HID_F);
    float x = h[i] + b[f];
    x = (x - m[f]) * rsqrtf(v[f] + EPS_BN) * g[f] + beta[f];
    h[i] = fmaxf(x, 0.0f);
}

// als[n,h] = <Hh[n,h,:], a_src[h,:]> ; ald likewise
__global__ void attn_logits(const float* __restrict__ Hh, const float* __restrict__ a_s,
                            const float* __restrict__ a_d, float* __restrict__ als,
                            float* __restrict__ ald, int F) {
    long i = (long)blockIdx.x * blockDim.x + threadIdx.x;
    if (i >= (long)N_NODES * HEADS) return;
    int n = (int)(i / HEADS), h = (int)(i % HEADS);
    const float* row = Hh + ((size_t)n * HEADS + h) * F;
    float ss = 0.f, sd = 0.f;
    for (int f = 0; f < F; ++f) {
        ss += row[f] * a_s[h * F + f];
        sd += row[f] * a_d[h * F + f];
    }
    als[i] = ss; ald[i] = sd;
}

__device__ __forceinline__ float leaky(float x) {
    return x > 0.f ? x : NEG_SLOPE * x;
}

// fp32 atomic max via sign-split int/uint trick (monotone in the max lattice)
__device__ __forceinline__ void atomicMaxF(float* addr, float val) {
    if (val >= 0.f) atomicMax((int*)addr, __float_as_int(val));
    else            atomicMin((unsigned int*)addr, __float_as_uint(val));
}

__global__ void edge_max(const int* __restrict__ src, const int* __restrict__ dst,
                         const float* __restrict__ als, const float* __restrict__ ald,
                         float* __restrict__ emax) {
    long idx = (long)blockIdx.x * blockDim.x + threadIdx.x;
    long e = idx / HEADS;
    if (e >= ETOT) return;
    int h = (int)(idx % HEADS);
    int s, d;
    if (e < N_EDGES) { s = src[e]; d = dst[e]; } else { s = d = (int)(e - N_EDGES); }
    float x = leaky(als[(size_t)s * HEADS + h] + ald[(size_t)d * HEADS + h]);
    atomicMaxF(&emax[(size_t)d * HEADS + h], x);
}

__global__ void edge_expsum(const int* __restrict__ src, const int* __restrict__ dst,
                            const float* __restrict__ als, const float* __restrict__ ald,
                            const float* __restrict__ emax, float* __restrict__ esum) {
    long idx = (long)blockIdx.x * blockDim.x + threadIdx.x;
    long e = idx / HEADS;
    if (e >= ETOT) return;
    int h = (int)(idx % HEADS);
    int s, d;
    if (e < N_EDGES) { s = src[e]; d = dst[e]; } else { s = d = (int)(e - N_EDGES); }
    float x = leaky(als[(size_t)s * HEADS + h] + ald[(size_t)d * HEADS + h]);
    atomicAdd(&esum[(size_t)d * HEADS + h], __expf(x - emax[(size_t)d * HEADS + h]));
}

// out[dst,h,:] += alpha * Hh[src,h,:]
__global__ void edge_aggr(const int* __restrict__ src, const int* __restrict__ dst,
                          const float* __restrict__ als, const float* __restrict__ ald,
                          const float* __restrict__ emax, const float* __restrict__ esum,
                          const float* __restrict__ Hh, float* __restrict__ out, int F) {
    int groups = F >> 2;
    long idx = (long)blockIdx.x * blockDim.x + threadIdx.x;
    long e = idx / (HEADS * groups);
    if (e >= ETOT) return;
    int rem = (int)(idx % (HEADS * groups));
    int h = rem / groups;
    int fg = (rem % groups) << 2;
    int s, d;
    if (e < N_EDGES) { s = src[e]; d = dst[e]; } else { s = d = (int)(e - N_EDGES); }
    float x = leaky(als[(size_t)s * HEADS + h] + ald[(size_t)d * HEADS + h]);
    float alpha = __expf(x - emax[(size_t)d * HEADS + h]) / esum[(size_t)d * HEADS + h];
    const float* hs = Hh + ((size_t)s * HEADS + h) * F + fg;
    float* od = out + ((size_t)d * HEADS + h) * F + fg;
    atomicAdd(&od[0], alpha * hs[0]);
    atomicAdd(&od[1], alpha * hs[1]);
    atomicAdd(&od[2], alpha * hs[2]);
    atomicAdd(&od[3], alpha * hs[3]);
}

__global__ void head_mean_bias_relu(const float* __restrict__ o, const float* __restrict__ b,
                                    float* __restrict__ out, int F) {
    long i = (long)blockIdx.x * blockDim.x + threadIdx.x;
    if (i >= (long)N_NODES * F) return;
    int n = (int)(i / F), f = (int)(i % F);
    float x = 0.5f * (o[(size_t)n * 2 * F + f] + o[(size_t)n * 2 * F + F + f]) + b[f];
    out[i] = fmaxf(x, 0.0f);
}

__global__ void gat2_logsoftmax(const float* __restrict__ o, const float* __restrict__ b,
                                float* __restrict__ out) {
    int n = blockIdx.x * blockDim.x + threadIdx.x;
    if (n >= N_NODES) return;
    float r[OUT_F];
    float mx = -1e30f;
    #pragma unroll
    for (int f = 0; f < OUT_F; ++f) {
        r[f] = 0.5f * (o[(size_t)n * 2 * OUT_F + f] + o[(size_t)n * 2 * OUT_F + OUT_F + f]) + b[f];
        mx = fmaxf(mx, r[f]);
    }
    float s = 0.f;
    #pragma unroll
    for (int f = 0; f < OUT_F; ++f) s += __expf(r[f] - mx);
    float lse = __logf(s);
    #pragma unroll
    for (int f = 0; f < OUT_F; ++f) out[(size_t)n * OUT_F + f] = r[f] - mx - lse;
}

// ---------------------------------------------------------------------------
static inline int nblk(long n) { return (int)((n + 255) / 256); }
static inline int gemm_blk() {
    long waves = (long)(N_NODES + 15) / 16;          // one wave per 16-row stripe
    return (int)((waves * 32 + 255) / 256);
}

extern "C" void kernel_launch(void* const* d_in, const int* in_sizes, int n_in,
                              void* d_out, int out_size, void* d_ws, size_t ws_size,
                              hipStream_t stream) {
    const float* x       = (const float*)d_in[0];
    const int*   ei      = (const int*)d_in[1];
    const int*   src     = ei;
    const int*   dst     = ei + N_EDGES;
    const float* gcn1_W  = (const float*)d_in[2];
    const float* gcn1_b  = (const float*)d_in[3];
    const float* bn1_g   = (const float*)d_in[4];
    const float* bn1_b   = (const float*)d_in[5];
    const float* bn1_m   = (const float*)d_in[6];
    const float* bn1_v   = (const float*)d_in[7];
    const float* gat1_W  = (const float*)d_in[8];
    const float* gat1_as = (const float*)d_in[9];
    const float* gat1_ad = (const float*)d_in[10];
    const float* gat1_b  = (const float*)d_in[11];
    const float* gcn2_W  = (const float*)d_in[12];
    const float* gcn2_b  = (const float*)d_in[13];
    const float* bn2_g   = (const float*)d_in[14];
    const float* bn2_b   = (const float*)d_in[15];
    const float* bn2_m   = (const float*)d_in[16];
    const float* bn2_v   = (const float*)d_in[17];
    const float* gat2_W  = (const float*)d_in[18];
    const float* gat2_as = (const float*)d_in[19];
    const float* gat2_ad = (const float*)d_in[20];
    const float* gat2_b  = (const float*)d_in[21];

    float* ws   = (float*)d_ws;
    float* dinv = ws;                                  // N
    float* bufA = ws + N_NODES;                        // N*64
    float* bufB = bufA + (size_t)N_NODES * 64;         // N*64
    float* bufC = bufB + (size_t)N_NODES * 64;         // N*64
    float* als  = bufC + (size_t)N_NODES * 64;         // N*2
    float* ald  = als + (size_t)N_NODES * 2;
    float* emax = ald + (size_t)N_NODES * 2;
    float* esum = emax + (size_t)N_NODES * 2;

    // ---- symmetric-norm degrees (self loop counted via init=1) ----
    fill_kernel<<<nblk(N_NODES), 256, 0, stream>>>(dinv, 1.0f, N_NODES);
    deg_kernel<<<nblk(N_EDGES), 256, 0, stream>>>(dst, dinv);
    rsqrt_kernel<<<nblk(N_NODES), 256, 0, stream>>>(dinv);

    // ---- GCN1: h = X @ W1 ; scatter ; bias+BN+ReLU ----
    wmma_gemm<IN_F, HID_F><<<gemm_blk(), 256, 0, stream>>>(x, gcn1_W, bufA);
    fill_kernel<<<nblk((long)N_NODES * HID_F), 256, 0, stream>>>(bufB, 0.f, (long)N_NODES * HID_F);
    gcn_scatter<<<nblk((long)ETOT * 8), 256, 0, stream>>>(bufA, src, dst, dinv, bufB, HID_F);
    bias_bn_relu<<<nblk((long)N_NODES * HID_F), 256, 0, stream>>>(bufB, gcn1_b, bn1_g, bn1_b, bn1_m, bn1_v);

    // ---- GAT1 (32 -> 2x32, mean heads) ----
    wmma_gemm<HID_F, HEADS * HID_F><<<gemm_blk(), 256, 0, stream>>>(bufB, gat1_W, bufC);
    attn_logits<<<nblk((long)N_NODES * HEADS), 256, 0, stream>>>(bufC, gat1_as, gat1_ad, als, ald, HID_F);
    fill_kernel<<<nblk((long)N_NODES * HEADS), 256, 0, stream>>>(emax, -1e30f, (long)N_NODES * HEADS);
    edge_max<<<nblk((long)ETOT * HEADS), 256, 0, stream>>>(src, dst, als, ald, emax);
    fill_kernel<<<nblk((long)N_NODES * HEADS), 256, 0, stream>>>(esum, 0.f, (long)N_NODES * HEADS);
    edge_expsum<<<nblk((long)ETOT * HEADS), 256, 0, stream>>>(src, dst, als, ald, emax, esum);
    fill_kernel<<<nblk((long)N_NODES * 64), 256, 0, stream>>>(bufA, 0.f, (long)N_NODES * 64);
    edge_aggr<<<nblk((long)ETOT * HEADS * 8), 256, 0, stream>>>(src, dst, als, ald, emax, esum, bufC, bufA, HID_F);
    head_mean_bias_relu<<<nblk((long)N_NODES * HID_F), 256, 0, stream>>>(bufA, gat1_b, bufB, HID_F);

    // ---- GCN2 ----
    wmma_gemm<HID_F, HID_F><<<gemm_blk(), 256, 0, stream>>>(bufB, gcn2_W, bufA);
    fill_kernel<<<nblk((long)N_NODES * HID_F), 256, 0, stream>>>(bufC, 0.f, (long)N_NODES * HID_F);
    gcn_scatter<<<nblk((long)ETOT * 8), 256, 0, stream>>>(bufA, src, dst, dinv, bufC, HID_F);
    bias_bn_relu<<<nblk((long)N_NODES * HID_F), 256, 0, stream>>>(bufC, gcn2_b, bn2_g, bn2_b, bn2_m, bn2_v);

    // ---- GAT2 (32 -> 2x16, mean heads) + log_softmax ----
    wmma_gemm<HID_F, HEADS * OUT_F><<<gemm_blk(), 256, 0, stream>>>(bufC, gat2_W, bufA);
    attn_logits<<<nblk((long)N_NODES * HEADS), 256, 0, stream>>>(bufA, gat2_as, gat2_ad, als, ald, OUT_F);
    fill_kernel<<<nblk((long)N_NODES * HEADS), 256, 0, stream>>>(emax, -1e30f, (long)N_NODES * HEADS);
    edge_max<<<nblk((long)ETOT * HEADS), 256, 0, stream>>>(src, dst, als, ald, emax);
    fill_kernel<<<nblk((long)N_NODES * HEADS), 256, 0, stream>>>(esum, 0.f, (long)N_NODES * HEADS);
    edge_expsum<<<nblk((long)ETOT * HEADS), 256, 0, stream>>>(src, dst, als, ald, emax, esum);
    fill_kernel<<<nblk((long)N_NODES * HEADS * OUT_F), 256, 0, stream>>>(bufB, 0.f, (long)N_NODES * HEADS * OUT_F);
    edge_aggr<<<nblk((long)ETOT * HEADS * 4), 256, 0, stream>>>(src, dst, als, ald, emax, esum, bufA, bufB, OUT_F);
    gat2_logsoftmax<<<nblk(N_NODES), 256, 0, stream>>>(bufB, gat2_b, (float*)d_out);
}